// EntropyFunctional_40587440947652
// MI455X (gfx1250) — compile-verified
//
#include <hip/hip_runtime.h>

// ---------------------------------------------------------------------------
// Problem constants (from reference setup_inputs): BSZ=32, DIM=2048, cg_iters=10
// Output: single fp32 scalar  -mean_b( <x_cg_b, H_b v_b> )
// ---------------------------------------------------------------------------
#define BSZ      32
#define DIM      2048
#define CG_ITERS 10
#define CG_ATOL  1e-3f

#define TILE_ROWS 128      // rows of H per block (8 waves x 16)
#define TILE_K    64       // K columns staged per LDS tile
#define LDSW      68       // padded row stride (floats): 272B = 17*16B, banks 4r mod 64
#define NTILES    (DIM / TILE_K)   // 32

typedef __attribute__((ext_vector_type(2))) float v2f;
typedef __attribute__((ext_vector_type(8))) float v8f;

// Async global->LDS 16B copy (ASYNCcnt). Inline asm: portable across the two
// toolchains' differing builtin arities (see bridge doc). Operand order per
// VGLOBAL encoding: vdst = LDS byte address VGPR, vaddr = 64-bit global addr.
__device__ __forceinline__ void async_g2l_b128(unsigned lds_off, const float* g) {
    asm volatile("global_load_async_to_lds_b128 %0, %1, off"
                 :: "v"(lds_off), "v"(g) : "memory");
}
__device__ __forceinline__ void wait_async0() {
    asm volatile("s_wait_asynccnt 0x0" ::: "memory");
}
__device__ __forceinline__ unsigned lds_off_of(const void* p) {
    // generic pointer to __shared__ has the LDS byte offset in its low 32 bits
    return (unsigned)(size_t)p;
}

// ---------------------------------------------------------------------------
// Batched GEMV  out[b] = H[b] * p[b]  using V_WMMA_F32_16X16X4_F32 with
// double-buffered async global->LDS staging of 128x64 H tiles.
//
// A-frag (ISA 7.12.2, 32-bit A 16x4): lane L<16 -> row M=L, K=0,1; lane L+16
// -> row M=L, K=2,3.  B = 4x16 with p in column N=0 (lanes 0/16), zeros
// elsewhere — free, since the kernel is HBM-bound by ~100x.  D column N=0 is
// lane 0 (M=0..7) and lane 16 (M=8..15).
// ---------------------------------------------------------------------------
__global__ __launch_bounds__(256) void cg_matvec_wmma(
    const float* __restrict__ Hmat,   // [BSZ, DIM, DIM]
    const float* __restrict__ pvec,   // [BSZ, DIM]
    float* __restrict__ out)          // [BSZ, DIM]
{
    __shared__ __align__(16) float lds_tile[2][TILE_ROWS * LDSW];
    __shared__ __align__(16) float lds_p[DIM];

    const int bi    = blockIdx.x >> 4;      // batch (grid = BSZ * 16)
    const int chunk = blockIdx.x & 15;      // 128-row chunk within batch
    const int t     = threadIdx.x;
    const int wave  = t >> 5;
    const int lane  = t & 31;
    const int m     = lane & 15;            // row within 16-row tile
    const int khalf = lane >> 4;            // 0: K=0,1   1: K=2,3
    const int koff  = khalf << 1;

    const int rowBase = chunk * TILE_ROWS;
    const float* __restrict__ Hbase = Hmat + (size_t)bi * DIM * DIM;
    const float* __restrict__ pb    = pvec + (size_t)bi * DIM;

    // ---- stage p[b] into LDS (2048 floats = 512 x b128, 2 per thread) ----
    const unsigned p_off = lds_off_of(&lds_p[0]);
#pragma unroll
    for (int q = 0; q < 2; ++q) {
        const int idx = (q * 256 + t) * 4;              // float index, 16B chunks
        async_g2l_b128(p_off + idx * 4, pb + idx);
    }

    // ---- per-thread tile-load mapping: 8 x b128 per 128x64 tile ----------
    // chunkid = it*256 + t ; row = chunkid>>4 ; c16 = chunkid&15
    const unsigned tile_off[2] = { lds_off_of(&lds_tile[0][0]),
                                   lds_off_of(&lds_tile[1][0]) };
    const int ldRow = t >> 4;          // base row for this thread (advances by 16)
    const int ldCol = (t & 15) * 4;    // float column of this thread's 16B chunk

    // issue tile 0
#pragma unroll
    for (int s = 0; s < 8; ++s) {
        const int row = ldRow + s * 16;
        async_g2l_b128(tile_off[0] + (row * LDSW + ldCol) * 4,
                       Hbase + (size_t)(rowBase + row) * DIM + ldCol);
    }

    const float bmask = (m == 0) ? 1.0f : 0.0f;   // B: only column N=0 nonzero
    v8f acc0 = {};    // even K-chunks
    v8f acc1 = {};    // odd  K-chunks (break the D->C dependence chain)

    for (int k = 0; k < NTILES; ++k) {
        wait_async0();       // this wave's tile-k (and p) asyncs have landed
        __syncthreads();     // everyone's landed; everyone done reading buf (k+1)%2

        if (k + 1 < NTILES) {
            const int jc = (k + 1) * TILE_K;
            const unsigned dst = tile_off[(k + 1) & 1];
#pragma unroll
            for (int s = 0; s < 8; ++s) {
                const int row = ldRow + s * 16;
                async_g2l_b128(dst + (row * LDSW + ldCol) * 4,
                               Hbase + (size_t)(rowBase + row) * DIM + jc + ldCol);
            }
        }

        // ---- compute tile k from LDS: 16 WMMAs over K = 64 ----------------
        const float* __restrict__ buf = &lds_tile[k & 1][0];
        const float* __restrict__ arow = buf + (wave * 16 + m) * LDSW + koff;
        const float* __restrict__ prow = &lds_p[k * TILE_K + koff];
#pragma unroll
        for (int u = 0; u < 16; u += 2) {
            v2f a0  = *(const v2f*)(arow + u * 4);       // ds_load_b64, conflict-free
            v2f a1  = *(const v2f*)(arow + u * 4 + 4);
            v2f pv0 = *(const v2f*)(prow + u * 4);       // broadcast read
            v2f pv1 = *(const v2f*)(prow + u * 4 + 4);
            v2f b0 = pv0 * bmask;
            v2f b1 = pv1 * bmask;
            acc0 = __builtin_amdgcn_wmma_f32_16x16x4_f32(
                       false, a0, false, b0, (short)0, acc0, false, false);
            acc1 = __builtin_amdgcn_wmma_f32_16x16x4_f32(
                       false, a1, false, b1, (short)0, acc1, false, false);
        }
    }

    v8f tot = acc0 + acc1;
    if (m == 0) {   // lanes 0 and 16 hold column N=0 of D
        float* o = out + (size_t)bi * DIM + rowBase + wave * 16 + (khalf << 3);
#pragma unroll
        for (int g = 0; g < 8; ++g) o[g] = tot[g];
    }
}

// ---------------------------------------------------------------------------
// Deterministic block tree reduction (256 threads).
// ---------------------------------------------------------------------------
__device__ __forceinline__ float block_reduce_sum(float v, float* sm) {
    const int t = threadIdx.x;
    sm[t] = v;
    __syncthreads();
#pragma unroll
    for (int s = 128; s > 0; s >>= 1) {
        if (t < s) sm[t] += sm[t + s];
        __syncthreads();
    }
    const float r = sm[0];
    __syncthreads();
    return r;
}

// x = 0, r = p = v, rs = sum(v*v).  One block per batch.
__global__ __launch_bounds__(256) void cg_init(
    const float* __restrict__ v,
    float* __restrict__ x, float* __restrict__ r,
    float* __restrict__ p, float* __restrict__ rs)
{
    __shared__ float sm[256];
    const int b = blockIdx.x;
    const float* vb = v + (size_t)b * DIM;
    float* xb = x + (size_t)b * DIM;
    float* rb = r + (size_t)b * DIM;
    float* pb = p + (size_t)b * DIM;
    float acc = 0.0f;
    for (int i = threadIdx.x; i < DIM; i += 256) {
        const float vi = vb[i];
        xb[i] = 0.0f; rb[i] = vi; pb[i] = vi;
        acc += vi * vi;
    }
    const float tot = block_reduce_sum(acc, sm);
    if (threadIdx.x == 0) rs[b] = tot;
}

// One CG step (mirrors the reference's masked update exactly). One block/batch.
__global__ __launch_bounds__(256) void cg_update(
    float* __restrict__ x, float* __restrict__ r,
    float* __restrict__ p, const float* __restrict__ Ap,
    float* __restrict__ rs)
{
    __shared__ float sm[256];
    const int b = blockIdx.x;
    float* xb = x + (size_t)b * DIM;
    float* rb = r + (size_t)b * DIM;
    float* pb = p + (size_t)b * DIM;
    const float* ab = Ap + (size_t)b * DIM;

    float accA = 0.0f;
    for (int i = threadIdx.x; i < DIM; i += 256) accA += pb[i] * ab[i];
    const float pAp = block_reduce_sum(accA, sm);

    const float rs_old = rs[b];
    const float alpha  = rs_old / fmaxf(pAp, 1e-30f);
    const bool  active = sqrtf(rs_old) > CG_ATOL;

    float accR = 0.0f;
    for (int i = threadIdx.x; i < DIM; i += 256) {
        const float pi = pb[i], ri = rb[i], ai = ab[i], xi = xb[i];
        const float xn = active ? (xi + alpha * pi) : xi;
        const float rn = active ? (ri - alpha * ai) : ri;
        xb[i] = xn; rb[i] = rn;
        accR += rn * rn;
    }
    const float rs_new = block_reduce_sum(accR, sm);
    const float beta   = rs_new / fmaxf(rs_old, 1e-30f);

    for (int i = threadIdx.x; i < DIM; i += 256) {
        const float pn = active ? (rb[i] + beta * pb[i]) : pb[i];
        pb[i] = pn;
    }
    if (threadIdx.x == 0) rs[b] = active ? rs_new : rs_old;
}

// value = -(1/BSZ) * sum_{b,i} x[b][i] * Hv[b][i].  Single block, fixed order.
__global__ __launch_bounds__(256) void cg_finalize(
    const float* __restrict__ x, const float* __restrict__ Hv,
    float* __restrict__ out)
{
    __shared__ float sm[256];
    float acc = 0.0f;
    for (int i = threadIdx.x; i < BSZ * DIM; i += 256) acc += x[i] * Hv[i];
    const float tot = block_reduce_sum(acc, sm);
    if (threadIdx.x == 0) out[0] = -tot / (float)BSZ;
}

// ---------------------------------------------------------------------------
// d_in order: x [32*2048] (unused), v [32*2048], H [32*2048*2048], cg_iters [1]
// (cg_iters is fixed at 10 by setup; a graph-captured launch sequence must be
// static, so the 10 iterations are unrolled as kernel launches.)
// ---------------------------------------------------------------------------
extern "C" void kernel_launch(void* const* d_in, const int* in_sizes, int n_in,
                              void* d_out, int out_size, void* d_ws, size_t ws_size,
                              hipStream_t stream) {
    (void)in_sizes; (void)n_in; (void)out_size; (void)ws_size;
    const float* v = (const float*)d_in[1];
    const float* H = (const float*)d_in[2];
    float* out = (float*)d_out;

    float* ws = (float*)d_ws;
    float* X  = ws;                        // [BSZ*DIM]
    float* R  = ws + 1 * BSZ * DIM;        // [BSZ*DIM]
    float* P  = ws + 2 * BSZ * DIM;        // [BSZ*DIM]
    float* AP = ws + 3 * BSZ * DIM;        // [BSZ*DIM]
    float* RS = ws + 4 * BSZ * DIM;        // [BSZ]

    const dim3 mvGrid(BSZ * (DIM / TILE_ROWS));  // 512 blocks

    cg_init<<<BSZ, 256, 0, stream>>>(v, X, R, P, RS);

    for (int it = 0; it < CG_ITERS; ++it) {
        cg_matvec_wmma<<<mvGrid, 256, 0, stream>>>(H, P, AP);
        cg_update<<<BSZ, 256, 0, stream>>>(X, R, P, AP, RS);
    }

    // Hv = H v  (same values as the detached H in the reference)
    cg_matvec_wmma<<<mvGrid, 256, 0, stream>>>(H, v, AP);
    cg_finalize<<<1, 256, 0, stream>>>(X, AP, out);
}